// GraphKMeans_87222195847761
// MI455X (gfx1250) — compile-verified
//
#include <hip/hip_runtime.h>
#include <cstdint>

// ---------------------------------------------------------------------------
// Soft k-means loss, MI455X (gfx1250, wave32).
//   dist[k,n] = ||x_n||^2 + ||c_k||^2 - 2 * (c . x)
//   loss = LAMBD * mean_n( sum_k dist * softmax_k(-alpha*dist) )
// GEMM via V_WMMA_F32_16X16X4_F32: A = centroids (M=clusters), B = x (N=points)
// ---------------------------------------------------------------------------

typedef float v2f __attribute__((ext_vector_type(2)));
typedef float v8f __attribute__((ext_vector_type(8)));
typedef int   v2i_gcc __attribute__((vector_size(8)));   // matches builtin param

#define LAMBD_F 0.1f

constexpr int kN = 262144;          // points
constexpr int kD = 512;             // embed dim
constexpr int kK = 64;              // clusters
constexpr int kWavesPerBlock = 8;
constexpr int kBlockThreads  = 256; // 8 x wave32
constexpr int kPtsPerWave    = 16;
constexpr int kTiles  = kN / kPtsPerWave;          // 16384
constexpr int kBlocks = kTiles / kWavesPerBlock;   // 2048
constexpr int kChunks = kD / 4;                    // 128 K-chunks of 4
constexpr int kAPairs = kK * kD / 2;               // 16384 float2 = 128 KiB

// ---- CDNA5 async global->LDS path (guarded; sync fallback) ----------------
#if defined(__has_builtin)
#  if __has_builtin(__builtin_amdgcn_global_load_async_to_lds_b64)
#    define HAVE_ASYNC_LDS 1
#  endif
#  if __has_builtin(__builtin_amdgcn_s_wait_asynccnt)
#    define HAVE_WAIT_ASYNC 1
#  endif
#endif

typedef __attribute__((address_space(1))) v2i_gcc* gptr64_t;   // global (AS1)
typedef __attribute__((address_space(3))) v2i_gcc* lptr64_t;   // LDS    (AS3)

__device__ __forceinline__ gptr64_t as_global64(const void* p) {
  return (gptr64_t)(uintptr_t)p;
}
__device__ __forceinline__ lptr64_t as_lds64(void* p) {
  // flat shared addr: low 32 bits are the LDS offset (ISA 10.2 aperture rules)
  return (lptr64_t)(uint32_t)(uintptr_t)p;
}

__global__ __launch_bounds__(kBlockThreads)
void kmeans_soft_loss(const float* __restrict__ x,
                      const float* __restrict__ cent,
                      const int*   __restrict__ alpha_p,
                      float*       __restrict__ out)
{
  extern __shared__ float smem[];
  float* afrag = smem;                   // kAPairs float2, swizzled A-fragments
  float* c2    = smem + 2 * kAPairs;     // 64 centroid squared norms
  float* sacc  = c2 + kK;                // block accumulator

  const int tid   = threadIdx.x;
  const int lane  = tid & 31;
  const int wave  = tid >> 5;
  const int halfw = lane >> 4;           // 0: K%4 in {0,1}, 1: K%4 in {2,3}

  if (tid == 0) *sacc = 0.0f;

  // ---- Stage centroids into LDS, pre-swizzled into WMMA A-fragment order.
  // pair p covers c[crow][2j], c[crow][2j+1]; dest lane = m + 16*(j&1),
  // chunk = j>>1, tile t = crow>>4  =>  slot index == consuming lane.
  #pragma unroll 4
  for (int i = 0; i < kAPairs / kBlockThreads; ++i) {        // 64 iters
    const int p    = tid + i * kBlockThreads;
    const int crow = p >> 8;                                  // 0..63
    const int j    = p & 255;
    const int t    = crow >> 4;
    const int m    = crow & 15;
    const int slot = ((t * kChunks + (j >> 1)) << 5) + m + ((j & 1) << 4);
    const float* g = cent + 2 * p;                            // coalesced
#if defined(HAVE_ASYNC_LDS)
    __builtin_amdgcn_global_load_async_to_lds_b64(
        as_global64((const void*)g), as_lds64((void*)(afrag + 2 * slot)), 0, 0);
#else
    *(float2*)(afrag + 2 * slot) = *(const float2*)g;
#endif
  }

  // ---- centroid squared norms (centroids are L2-hot; one-time per block)
  if (tid < kK) {
    float s = 0.0f;
    const float* cr = cent + tid * kD;
    #pragma unroll 8
    for (int k = 0; k < kD; ++k) s = fmaf(cr[k], cr[k], s);
    c2[tid] = s;
  }

#if defined(HAVE_ASYNC_LDS)
#  if defined(HAVE_WAIT_ASYNC)
  __builtin_amdgcn_s_wait_asynccnt(0);
#  else
  asm volatile("s_wait_asynccnt 0" ::: "memory");
#  endif
#endif
  __syncthreads();

  // ---- streaming GEMM: this wave's 16 points vs all 64 centroids ----------
  const int    tile    = blockIdx.x * kWavesPerBlock + wave;
  const size_t rowbase = (size_t)tile * kPtsPerWave;
  const int    n       = lane & 15;                 // point column

  const float2* xpairs = (const float2*)(x + (rowbase + (size_t)n) * kD);
  const float2* afrag2 = ((const float2*)afrag) + lane;   // conflict-free

  v8f acc[4] = {};       // 4 M-tiles of clusters (16 each)
  float x2p = 0.0f;      // partial ||x_n||^2 (this lane's K residues)

  #pragma unroll 4
  for (int c = 0; c < kChunks; ++c) {
    const float2 bv = xpairs[2 * c + halfw];        // x[n][kbase..kbase+1]
    x2p = fmaf(bv.x, bv.x, fmaf(bv.y, bv.y, x2p));
    v2f b; b[0] = bv.x; b[1] = bv.y;
    #pragma unroll
    for (int t = 0; t < 4; ++t) {
      const float2 av = afrag2[(t * kChunks + c) << 5];   // ds_load_b64
      v2f a; a[0] = av.x; a[1] = av.y;
      acc[t] = __builtin_amdgcn_wmma_f32_16x16x4_f32(
          false, a, false, b, (short)0, acc[t], false, false);
    }
  }

  // ---- fused softmax + loss. Lane owns point n and 32 of its 64 cluster
  // values; lane^16 owns the other 32 -> single xor-16 combines everything.
  const float alpha = (float)alpha_p[0];
  const float x2f = x2p + __shfl_xor(x2p, 16, 32);  // full ||x_n||^2

  float dist[4][8];
  float mn = 3.402823466e38f;
  #pragma unroll
  for (int t = 0; t < 4; ++t) {
    #pragma unroll
    for (int j = 0; j < 8; ++j) {
      // cluster = 16*t + 8*halfw + j  (C/D VGPR layout)
      const float dd = x2f + c2[t * 16 + halfw * 8 + j] - 2.0f * acc[t][j];
      dist[t][j] = dd;
      mn = fminf(mn, dd);
    }
  }
  mn = fminf(mn, __shfl_xor(mn, 16, 32));

  float s1 = 0.0f, s2 = 0.0f;
  #pragma unroll
  for (int t = 0; t < 4; ++t) {
    #pragma unroll
    for (int j = 0; j < 8; ++j) {
      const float dd = dist[t][j];
      const float e  = __expf(alpha * (mn - dd));   // softmax(-alpha*dist)
      s1 += e;
      s2  = fmaf(dd, e, s2);
    }
  }
  s1 += __shfl_xor(s1, 16, 32);
  s2 += __shfl_xor(s2, 16, 32);

  float contrib = (lane < 16) ? (s2 / s1) : 0.0f;   // one point per low lane
  contrib += __shfl_xor(contrib, 1, 32);
  contrib += __shfl_xor(contrib, 2, 32);
  contrib += __shfl_xor(contrib, 4, 32);
  contrib += __shfl_xor(contrib, 8, 32);
  contrib += __shfl_xor(contrib, 16, 32);           // wave total (16 points)

  if (lane == 0) atomicAdd(sacc, contrib);
  __syncthreads();
  if (tid == 0) atomicAdd(out, *sacc * (LAMBD_F / (float)kN));
}

// ---------------------------------------------------------------------------
extern "C" void kernel_launch(void* const* d_in, const int* in_sizes, int n_in,
                              void* d_out, int out_size, void* d_ws, size_t ws_size,
                              hipStream_t stream) {
  (void)in_sizes; (void)n_in; (void)out_size; (void)d_ws; (void)ws_size;
  const float* x    = (const float*)d_in[0];
  const float* cent = (const float*)d_in[1];
  const int*   alph = (const int*)d_in[2];
  float* out = (float*)d_out;

  const size_t shmem = (size_t)(2 * kAPairs + kK + 4) * sizeof(float); // ~128.3 KiB

  // >64KB dynamic LDS opt-in (deterministic, idempotent; not a stream op)
  (void)hipFuncSetAttribute(reinterpret_cast<const void*>(kmeans_soft_loss),
                            hipFuncAttributeMaxDynamicSharedMemorySize, (int)shmem);

  (void)hipMemsetAsync(out, 0, sizeof(float), stream);    // graph-capturable
  kmeans_soft_loss<<<kBlocks, kBlockThreads, shmem, stream>>>(x, cent, alph, out);
}